// GraphNet_87488483820081
// MI455X (gfx1250) — compile-verified
//
#include <hip/hip_runtime.h>

// Problem constants (fixed by the reference)
#define NN 40000     // nodes
#define NE 640000    // edges
#define DD 128       // D_IN == H
#define RR 8         // relations
#define BB 30        // bases

typedef float v2f __attribute__((ext_vector_type(2)));
typedef float v8f __attribute__((ext_vector_type(8)));

// ---------------------------------------------------------------------------
// fp32 hardware atomic add (global_atomic_add_f32)
__device__ __forceinline__ void atomic_add_f32(float* p, float v) {
    unsafeAtomicAdd(p, v);
}

// ---------------------------------------------------------------------------
// 16x16 output tile, K=128 reduction with V_WMMA_F32_16X16X4_F32.
// A: [*,128] row-major, Wm: [128,128] row-major.
// Fragment layout per ISA 7.12.2:
//   A 16x4 f32: lanes 0-15 hold M=lane, K=koff..koff+1 with koff = (lane>=16)?2:0
//   B 4x16 f32: lanes 0-15 hold N=lane, same K split
__device__ __forceinline__ v8f wmma_tile_k128(const float* __restrict__ A,
                                              const float* __restrict__ Wm,
                                              int mrow, int ncol, int koff,
                                              v8f acc) {
    const float* a_ptr = A + (size_t)mrow * DD + koff;
    const float* b_ptr = Wm + (size_t)koff * DD + ncol;
#pragma unroll 8
    for (int k = 0; k < DD; k += 4) {
        v2f a = *(const v2f*)(a_ptr + k);
        v2f b;
        b[0] = b_ptr[k * DD];
        b[1] = b_ptr[k * DD + DD];
        acc = __builtin_amdgcn_wmma_f32_16x16x4_f32(
            /*neg_a=*/false, a, /*neg_b=*/false, b,
            /*c_mod=*/(short)0, acc, /*reuse_a=*/false, /*reuse_b=*/false);
    }
    return acc;
}

// ---------------------------------------------------------------------------
__global__ void zero_kernel(float* __restrict__ p, int n) {
    int i = blockIdx.x * blockDim.x + threadIdx.x;
    if (i < n) p[i] = 0.0f;
}

// w[r,d,o] = sum_b comp[r,b] * basis[b,d,o]   (R*128*128 outputs, trivial)
__global__ void compute_w_kernel(const float* __restrict__ comp,
                                 const float* __restrict__ basis,
                                 float* __restrict__ w) {
    int i  = blockIdx.x * blockDim.x + threadIdx.x;   // over R*128*128
    int r  = i >> 14;
    int dh = i & 16383;
    float acc = 0.0f;
#pragma unroll
    for (int b = 0; b < BB; ++b)
        acc = fmaf(comp[r * BB + b], basis[b * 16384 + dh], acc);
    w[i] = acc;
}

// xw[r] = X @ w[r]  for all r.  One 16x16 tile per wave.
// tiles = 2500(M) * 8(Ncols) * 8(R) = 160000 waves -> 20000 blocks x 8 waves
__global__ void __launch_bounds__(256)
xw_gemm_kernel(const float* __restrict__ X, const float* __restrict__ W,
               float* __restrict__ XW) {
    const int lane = threadIdx.x & 31;
    const int wave = threadIdx.x >> 5;
    const int tile = blockIdx.x * 8 + wave;
    const int r     = tile / 20000;
    const int rem   = tile % 20000;
    const int mtile = rem >> 3;
    const int ntile = rem & 7;

    const int mrow = (mtile << 4) + (lane & 15);
    const int ncol = (ntile << 4) + (lane & 15);
    const int koff = (lane >> 4) << 1;

    v8f acc = {};
    acc = wmma_tile_k128(X, W + r * 16384, mrow, ncol, koff, acc);

    // C/D layout: vgpr i -> row = i + 8*(lane>=16), col = lane&15
    float* out = XW + (size_t)r * NN * DD + (size_t)(mtile << 4) * DD + (ntile << 4);
    const int rbase = (lane >> 4) << 3;
    const int c = lane & 15;
#pragma unroll
    for (int i = 0; i < 8; ++i)
        out[(rbase + i) * DD + c] = acc[i];
}

// agg[dst] += xw[etype, src]; deg[dst] += 1.  One wave per edge.
__global__ void __launch_bounds__(256)
scatter_rgcn_kernel(const int* __restrict__ ei, const int* __restrict__ et,
                    const float* __restrict__ XW, float* __restrict__ agg,
                    float* __restrict__ deg) {
    const int lane = threadIdx.x & 31;
    const int e = blockIdx.x * 8 + (threadIdx.x >> 5);
    if (e >= NE) return;
    const int src = ei[e];
    const int dst = ei[NE + e];
    const int r   = et[e];
    const float4 v = *(const float4*)(XW + (size_t)r * NN * DD +
                                      (size_t)src * DD + lane * 4);
    float* ap = agg + (size_t)dst * DD + lane * 4;
    atomic_add_f32(ap + 0, v.x);
    atomic_add_f32(ap + 1, v.y);
    atomic_add_f32(ap + 2, v.z);
    atomic_add_f32(ap + 3, v.w);
    if (lane == 0) atomic_add_f32(deg + dst, 1.0f);
}

// h = agg / max(deg,1) + X @ root + bias1.  2500 blocks x 8 waves.
__global__ void __launch_bounds__(256)
h_kernel(const float* __restrict__ X, const float* __restrict__ root,
         const float* __restrict__ agg, const float* __restrict__ deg,
         const float* __restrict__ bias1, float* __restrict__ h) {
    const int lane = threadIdx.x & 31;
    const int tile = blockIdx.x * 8 + (threadIdx.x >> 5);
    const int mtile = tile >> 3;
    const int ntile = tile & 7;

    const int mrow = (mtile << 4) + (lane & 15);
    const int ncol = (ntile << 4) + (lane & 15);
    const int koff = (lane >> 4) << 1;

    v8f acc = {};
    acc = wmma_tile_k128(X, root, mrow, ncol, koff, acc);

    const int rbase = (lane >> 4) << 3;
    const int c = (ntile << 4) + (lane & 15);
    const float bc = bias1[c];
#pragma unroll
    for (int i = 0; i < 8; ++i) {
        const int row = (mtile << 4) + rbase + i;
        float d = deg[row];
        d = d < 1.0f ? 1.0f : d;
        h[(size_t)row * DD + c] = acc[i] + agg[(size_t)row * DD + c] / d + bc;
    }
}

// nbr[dst] += h[src].  One wave per edge.
__global__ void __launch_bounds__(256)
scatter_h_kernel(const int* __restrict__ ei, const float* __restrict__ h,
                 float* __restrict__ nbr) {
    const int lane = threadIdx.x & 31;
    const int e = blockIdx.x * 8 + (threadIdx.x >> 5);
    if (e >= NE) return;
    const int src = ei[e];
    const int dst = ei[NE + e];
    const float4 v = *(const float4*)(h + (size_t)src * DD + lane * 4);
    float* np = nbr + (size_t)dst * DD + lane * 4;
    atomic_add_f32(np + 0, v.x);
    atomic_add_f32(np + 1, v.y);
    atomic_add_f32(np + 2, v.z);
    atomic_add_f32(np + 3, v.w);
}

// out = nbr @ w_rel + h @ w_root2 + bias2.  2500 blocks x 8 waves.
__global__ void __launch_bounds__(256)
out_kernel(const float* __restrict__ nbr, const float* __restrict__ w_rel,
           const float* __restrict__ h, const float* __restrict__ w_root2,
           const float* __restrict__ bias2, float* __restrict__ out) {
    const int lane = threadIdx.x & 31;
    const int tile = blockIdx.x * 8 + (threadIdx.x >> 5);
    const int mtile = tile >> 3;
    const int ntile = tile & 7;

    const int mrow = (mtile << 4) + (lane & 15);
    const int ncol = (ntile << 4) + (lane & 15);
    const int koff = (lane >> 4) << 1;

    v8f acc = {};
    acc = wmma_tile_k128(nbr, w_rel, mrow, ncol, koff, acc);
    acc = wmma_tile_k128(h, w_root2, mrow, ncol, koff, acc);

    const int rbase = (lane >> 4) << 3;
    const int c = (ntile << 4) + (lane & 15);
    const float bc = bias2[c];
#pragma unroll
    for (int i = 0; i < 8; ++i) {
        const int row = (mtile << 4) + rbase + i;
        out[(size_t)row * DD + c] = acc[i] + bc;
    }
}

// ---------------------------------------------------------------------------
extern "C" void kernel_launch(void* const* d_in, const int* in_sizes, int n_in,
                              void* d_out, int out_size, void* d_ws, size_t ws_size,
                              hipStream_t stream) {
    (void)in_sizes; (void)n_in; (void)out_size; (void)ws_size;

    const float* X      = (const float*)d_in[0];   // [N,128]
    const int*   ei     = (const int*)d_in[1];     // [2,E]
    /* edge_norm d_in[2] unused by reference */
    const int*   et     = (const int*)d_in[3];     // [E]
    const float* basis  = (const float*)d_in[4];   // [B,128,128]
    const float* comp   = (const float*)d_in[5];   // [R,B]
    const float* root   = (const float*)d_in[6];   // [128,128]
    const float* bias1  = (const float*)d_in[7];   // [128]
    const float* w_rel  = (const float*)d_in[8];   // [128,128]
    const float* w_root2= (const float*)d_in[9];   // [128,128]
    const float* bias2  = (const float*)d_in[10];  // [128]
    float* out = (float*)d_out;                    // [N,128]

    // Workspace layout (floats)
    float* ws  = (float*)d_ws;
    float* w   = ws;                       // R*128*128      =   131072
    float* xw  = w   + 131072;             // R*N*128        = 40960000
    float* agg = xw  + (size_t)RR*NN*DD;   // N*128          =  5120000
    float* deg = agg + (size_t)NN*DD;      // N              =    40000
    float* nbr = deg + NN;                 // N*128          =  5120000
    float* h   = nbr + (size_t)NN*DD;      // N*128          =  5120000

    // zero agg + deg + nbr (contiguous)
    const int nzero = NN*DD + NN + NN*DD;  // 10,280,000
    zero_kernel<<<(nzero + 255) / 256, 256, 0, stream>>>(agg, nzero);

    compute_w_kernel<<<(RR*DD*DD) / 256, 256, 0, stream>>>(comp, basis, w);

    xw_gemm_kernel<<<20000, 256, 0, stream>>>(X, w, xw);

    scatter_rgcn_kernel<<<NE / 8, 256, 0, stream>>>(ei, et, xw, agg, deg);

    h_kernel<<<2500, 256, 0, stream>>>(X, root, agg, deg, bias1, h);

    scatter_h_kernel<<<NE / 8, 256, 0, stream>>>(ei, h, nbr);

    out_kernel<<<2500, 256, 0, stream>>>(nbr, w_rel, h, w_root2, bias2, out);
}